// AutoEncoder_21139829031562
// MI455X (gfx1250) — compile-verified
//
#include <hip/hip_runtime.h>

typedef __attribute__((ext_vector_type(2))) float v2f;
typedef __attribute__((ext_vector_type(8))) float v8f;

#define BN_EPS 1e-5f

// ---------------- direct conv 4x4, stride 2, pad 1 (OIHW weights) -----------
__global__ void vq_conv4x4s2(const float* __restrict__ x, const float* __restrict__ w,
                             const float* __restrict__ bias, float* __restrict__ y,
                             int N, int Cin, int Cout, int Hin, int Win,
                             int Hout, int Wout)
{
    int idx = blockIdx.x * blockDim.x + threadIdx.x;
    int total = N * Cout * Hout * Wout;
    if (idx >= total) return;
    int wo = idx % Wout; int t = idx / Wout;
    int ho = t % Hout;   t /= Hout;
    int co = t % Cout;   int n = t / Cout;
    float acc = bias[co];
    int hi0 = ho * 2 - 1, wi0 = wo * 2 - 1;
    for (int ci = 0; ci < Cin; ++ci) {
        const float* xp = x + ((n * Cin + ci) * Hin) * Win;
        const float* wp = w + (co * Cin + ci) * 16;
        #pragma unroll
        for (int kh = 0; kh < 4; ++kh) {
            int hi = hi0 + kh;
            if ((unsigned)hi >= (unsigned)Hin) continue;
            #pragma unroll
            for (int kw = 0; kw < 4; ++kw) {
                int wi = wi0 + kw;
                if ((unsigned)wi >= (unsigned)Win) continue;
                acc += xp[hi * Win + wi] * wp[kh * 4 + kw];
            }
        }
    }
    y[idx] = acc;
}

// ---------------- direct conv-transpose 4x4, stride 2, pad 1 (IOHW weights) -
__global__ void vq_convT4x4s2(const float* __restrict__ x, const float* __restrict__ w,
                              const float* __restrict__ bias, float* __restrict__ y,
                              int N, int Cin, int Cout, int Hin, int Win,
                              int Hout, int Wout)
{
    int idx = blockIdx.x * blockDim.x + threadIdx.x;
    int total = N * Cout * Hout * Wout;
    if (idx >= total) return;
    int wo = idx % Wout; int t = idx / Wout;
    int ho = t % Hout;   t /= Hout;
    int co = t % Cout;   int n = t / Cout;
    float acc = bias[co];
    for (int ci = 0; ci < Cin; ++ci) {
        const float* xp = x + ((n * Cin + ci) * Hin) * Win;
        const float* wp = w + (ci * Cout + co) * 16;
        #pragma unroll
        for (int kh = 0; kh < 4; ++kh) {
            int ht = ho + 1 - kh;
            if (ht & 1) continue;
            int hi = ht >> 1;
            if ((unsigned)hi >= (unsigned)Hin) continue;
            #pragma unroll
            for (int kw = 0; kw < 4; ++kw) {
                int wt = wo + 1 - kw;
                if (wt & 1) continue;
                int wi = wt >> 1;
                if ((unsigned)wi >= (unsigned)Win) continue;
                acc += xp[hi * Win + wi] * wp[kh * 4 + kw];
            }
        }
    }
    y[idx] = acc;
}

// ---------------- 1x1 conv ---------------------------------------------------
__global__ void vq_conv1x1(const float* __restrict__ x, const float* __restrict__ w,
                           const float* __restrict__ bias, float* __restrict__ y,
                           int N, int Cin, int Cout, int HW)
{
    int idx = blockIdx.x * blockDim.x + threadIdx.x;
    int total = N * Cout * HW;
    if (idx >= total) return;
    int p = idx % HW; int t = idx / HW;
    int co = t % Cout; int n = t / Cout;
    float acc = bias[co];
    const float* xp = x + (n * Cin) * HW + p;
    const float* wp = w + co * Cin;
    for (int ci = 0; ci < Cin; ++ci)
        acc += xp[ci * HW] * wp[ci];
    y[idx] = acc;
}

// ---------------- 3x3 conv, stride 1, pad 1 ---------------------------------
__global__ void vq_conv3x3s1(const float* __restrict__ x, const float* __restrict__ w,
                             const float* __restrict__ bias, float* __restrict__ y,
                             int N, int Cin, int Cout, int H, int W)
{
    int idx = blockIdx.x * blockDim.x + threadIdx.x;
    int total = N * Cout * H * W;
    if (idx >= total) return;
    int wo = idx % W; int t = idx / W;
    int ho = t % H;   t /= H;
    int co = t % Cout; int n = t / Cout;
    float acc = bias[co];
    for (int ci = 0; ci < Cin; ++ci) {
        const float* xp = x + ((n * Cin + ci) * H) * W;
        const float* wp = w + (co * Cin + ci) * 9;
        #pragma unroll
        for (int kh = 0; kh < 3; ++kh) {
            int hi = ho - 1 + kh;
            if ((unsigned)hi >= (unsigned)H) continue;
            #pragma unroll
            for (int kw = 0; kw < 3; ++kw) {
                int wi = wo - 1 + kw;
                if ((unsigned)wi >= (unsigned)W) continue;
                acc += xp[hi * W + wi] * wp[kh * 3 + kw];
            }
        }
    }
    y[idx] = acc;
}

// ---------------- BatchNorm: per-channel batch stats -> (scale, shift) ------
__global__ void vq_bn_stats(const float* __restrict__ x, const float* __restrict__ g,
                            const float* __restrict__ be, float* __restrict__ ss,
                            int N, int C, int HW)
{
    int c = blockIdx.x;
    int tid = threadIdx.x;
    long cnt = (long)N * HW;
    float s = 0.f, sq = 0.f;
    for (long i = tid; i < cnt; i += blockDim.x) {
        int n = (int)(i / HW), p = (int)(i % HW);
        float v = x[(long)(n * C + c) * HW + p];
        s += v; sq += v * v;
    }
    __shared__ float rs[256], rq[256];
    rs[tid] = s; rq[tid] = sq;
    __syncthreads();
    for (int o = 128; o > 0; o >>= 1) {
        if (tid < o) { rs[tid] += rs[tid + o]; rq[tid] += rq[tid + o]; }
        __syncthreads();
    }
    if (tid == 0) {
        float inv = 1.0f / (float)cnt;
        float mean = rs[0] * inv;
        float var = rq[0] * inv - mean * mean;
        float sc = g[c] * rsqrtf(var + BN_EPS);
        ss[c] = sc;
        ss[C + c] = be[c] - mean * sc;
    }
}

__global__ void vq_bn_apply(const float* __restrict__ x, const float* __restrict__ ss,
                            float* __restrict__ y, int total, int C, int HW, int relu)
{
    int idx = blockIdx.x * blockDim.x + threadIdx.x;
    if (idx >= total) return;
    int c = (idx / HW) % C;
    float v = x[idx] * ss[c] + ss[C + c];
    if (relu) v = fmaxf(v, 0.f);
    y[idx] = v;
}

// ---------------- ||e_k||^2 per codebook entry ------------------------------
__global__ void vq_enorm(const float* __restrict__ emb, float* __restrict__ en)
{
    int k = blockIdx.x * blockDim.x + threadIdx.x;
    if (k >= 1024) return;
    float s = 0.f;
    #pragma unroll
    for (int c = 0; c < 64; ++c) { float v = emb[k * 64 + c]; s += v * v; }
    en[k] = s;
}

// ---------------- VQ: fp32 WMMA GEMM + argmin + codebook gather -------------
// One workgroup = 64 spatial positions (4 M-tiles) x all 1024 codes.
// 256 threads = 8 waves. Per wave: 8 code tiles; for each, the 16 B fragments
// (codebook slice) are held in 32 VGPRs and reused across 4 M-tiles, with A
// re-read from LDS each M-tile (stride 66 -> 8B-aligned, bank-conflict-free).
// The rolled mt loop + compiler barrier stop LICM from promoting all 128 A
// values to registers (round-2 caused scratch spills in the WMMA loop).
__global__ void vq_assign_wmma(const float* __restrict__ ze,   // [32,64,56,56] BN'd
                               const float* __restrict__ emb,  // [1024,64]
                               const float* __restrict__ en,   // [1024]
                               float* __restrict__ zq)         // [32,64,56,56]
{
    __shared__ float zt[64 * 66];              // 64 positions x 64 ch, stride 66
    __shared__ unsigned long long mn[64];      // packed (ordered-score, code)
    __shared__ int codes[64];

    const int tid  = threadIdx.x;
    const int lane = tid & 31;
    const int wave = tid >> 5;                 // 0..7
    const int row  = lane & 15;
    const int half = lane >> 4;

    if (tid < 64) mn[tid] = ~0ULL;

    // stage Z tile: position m = blockIdx.x*64 + r ; Z[r][c] = ze[(n*64+c)*3136 + hw]
    for (int i = tid; i < 64 * 64; i += 256) {
        int r = i >> 6, c = i & 63;
        int m = blockIdx.x * 64 + r;
        int n = m / 3136, hw = m % 3136;
        zt[r * 66 + c] = ze[(n * 64 + c) * 3136 + hw];
    }
    __syncthreads();

    // 64 code tiles of 16; 8 waves -> 8 tiles each
    for (int t = wave; t < 64; t += 8) {
        int code = t * 16 + row;               // B: N = lane%16
        const float* ep = emb + code * 64;

        // keep this tile's 16 B fragments resident in VGPRs (reused 4x)
        float b0[16], b1[16];
        #pragma unroll
        for (int kk = 0; kk < 16; ++kk) {
            int ch = kk * 4 + half * 2;        // B: K = (lane/16)*2 + v
            b0[kk] = ep[ch];
            b1[kk] = ep[ch + 1];
        }
        float e2 = en[code];

        // opaque barrier: A must be re-read from LDS, not hoisted+spilled
        asm volatile("" ::: "memory");

        #pragma unroll 1
        for (int mt = 0; mt < 4; ++mt) {
            const float* ap = zt + (mt * 16 + row) * 66;   // A: M = lane%16
            v8f acc = {};
            #pragma unroll
            for (int kk = 0; kk < 16; ++kk) {
                int ch = kk * 4 + half * 2;    // A: K = (lane/16)*2 + v
                v2f a; a.x = ap[ch]; a.y = ap[ch + 1];
                v2f b; b.x = b0[kk]; b.y = b1[kk];
                acc = __builtin_amdgcn_wmma_f32_16x16x4_f32(
                          false, a, false, b, (short)0, acc, false, false);
            }
            #pragma unroll
            for (int v = 0; v < 8; ++v) {
                int M = mt * 16 + v + half * 8;            // C: M = v + 8*(lane/16)
                float score = e2 - 2.0f * acc[v];
                unsigned int u = __float_as_uint(score);
                u ^= (unsigned int)((int)u >> 31) | 0x80000000u;  // order-preserving
                unsigned long long key =
                    ((unsigned long long)u << 32) | (unsigned int)code;
                atomicMin(&mn[M], key);
            }
        }
    }
    __syncthreads();
    if (tid < 64) codes[tid] = (int)(mn[tid] & 0xFFFFFFFFull);
    __syncthreads();

    // gather z_q: 64 positions x 64 channels
    for (int i = tid; i < 64 * 64; i += 256) {
        int r = i >> 6, c = i & 63;
        int m = blockIdx.x * 64 + r;
        int n = m / 3136, hw = m % 3136;
        zq[(n * 64 + c) * 3136 + hw] = emb[codes[r] * 64 + c];
    }
}

// ---------------------------------------------------------------------------
extern "C" void kernel_launch(void* const* d_in, const int* in_sizes, int n_in,
                              void* d_out, int out_size, void* d_ws, size_t ws_size,
                              hipStream_t stream) {
    const float* x   = (const float*)d_in[0];
    const float* w1  = (const float*)d_in[1];
    const float* b1  = (const float*)d_in[2];
    const float* g1  = (const float*)d_in[3];
    const float* be1 = (const float*)d_in[4];
    const float* w2  = (const float*)d_in[5];
    const float* b2  = (const float*)d_in[6];
    const float* g2  = (const float*)d_in[7];
    const float* be2 = (const float*)d_in[8];
    const float* w3  = (const float*)d_in[9];
    const float* b3  = (const float*)d_in[10];
    const float* g3  = (const float*)d_in[11];
    const float* be3 = (const float*)d_in[12];
    const float* emb = (const float*)d_in[13];
    const float* dw1 = (const float*)d_in[14];
    const float* db1 = (const float*)d_in[15];
    const float* dg1 = (const float*)d_in[16];
    const float* dbe1= (const float*)d_in[17];
    const float* dw2 = (const float*)d_in[18];
    const float* db2 = (const float*)d_in[19];
    const float* dg2 = (const float*)d_in[20];
    const float* dbe2= (const float*)d_in[21];
    const float* w4  = (const float*)d_in[22];
    const float* b4  = (const float*)d_in[23];

    float* out = (float*)d_out;
    float* x_tilde = out;                      // 32*1*224*224   = 1605632
    float* z_e     = out + 1605632;            // 32*64*56*56    = 6422528
    float* z_q     = z_e + 6422528;            // 32*64*56*56

    float* ws = (float*)d_ws;
    size_t o = 0;
    float* t1 = ws + o; o += 6422528;          // [32,16,112,112]
    float* t2 = ws + o; o += 3211264;          // [32,32,56,56]
    float* t3 = ws + o; o += 6422528;          // [32,64,56,56] raw conv3
    float* t4 = ws + o; o += 12845056;         // [32,32,112,112]
    float* t5 = ws + o; o += 25690112;         // [32,16,224,224]
    float* st = ws + o; o += 128;              // per-layer (scale,shift), reused
    float* en = ws + o; o += 1024;             // ||e_k||^2

    const int TB = 256;
    const int N = 32;

    // encoder layer 1: conv 1->16, 224->112
    {
        int total = N * 16 * 112 * 112;
        vq_conv4x4s2<<<(total + TB - 1) / TB, TB, 0, stream>>>(
            x, w1, b1, t1, N, 1, 16, 224, 224, 112, 112);
        vq_bn_stats<<<16, TB, 0, stream>>>(t1, g1, be1, st, N, 16, 112 * 112);
        vq_bn_apply<<<(total + TB - 1) / TB, TB, 0, stream>>>(
            t1, st, t1, total, 16, 112 * 112, 1);
    }
    // encoder layer 2: conv 16->32, 112->56
    {
        int total = N * 32 * 56 * 56;
        vq_conv4x4s2<<<(total + TB - 1) / TB, TB, 0, stream>>>(
            t1, w2, b2, t2, N, 16, 32, 112, 112, 56, 56);
        vq_bn_stats<<<32, TB, 0, stream>>>(t2, g2, be2, st, N, 32, 56 * 56);
        vq_bn_apply<<<(total + TB - 1) / TB, TB, 0, stream>>>(
            t2, st, t2, total, 32, 56 * 56, 1);
    }
    // encoder layer 3: 1x1 conv 32->64, BN (no relu) -> z_e in d_out
    {
        int total = N * 64 * 56 * 56;
        vq_conv1x1<<<(total + TB - 1) / TB, TB, 0, stream>>>(
            t2, w3, b3, t3, N, 32, 64, 56 * 56);
        vq_bn_stats<<<64, TB, 0, stream>>>(t3, g3, be3, st, N, 64, 56 * 56);
        vq_bn_apply<<<(total + TB - 1) / TB, TB, 0, stream>>>(
            t3, st, z_e, total, 64, 56 * 56, 0);
    }
    // vector quantization: WMMA distance GEMM + argmin + gather -> z_q
    vq_enorm<<<4, TB, 0, stream>>>(emb, en);
    vq_assign_wmma<<<100352 / 64, TB, 0, stream>>>(z_e, emb, en, z_q);

    // decoder layer 1: convT 64->32, 56->112
    {
        int total = N * 32 * 112 * 112;
        vq_convT4x4s2<<<(total + TB - 1) / TB, TB, 0, stream>>>(
            z_q, dw1, db1, t4, N, 64, 32, 56, 56, 112, 112);
        vq_bn_stats<<<32, TB, 0, stream>>>(t4, dg1, dbe1, st, N, 32, 112 * 112);
        vq_bn_apply<<<(total + TB - 1) / TB, TB, 0, stream>>>(
            t4, st, t4, total, 32, 112 * 112, 1);
    }
    // decoder layer 2: convT 32->16, 112->224
    {
        int total = N * 16 * 224 * 224;
        vq_convT4x4s2<<<(total + TB - 1) / TB, TB, 0, stream>>>(
            t4, dw2, db2, t5, N, 32, 16, 112, 112, 224, 224);
        vq_bn_stats<<<16, TB, 0, stream>>>(t5, dg2, dbe2, st, N, 16, 224 * 224);
        vq_bn_apply<<<(total + TB - 1) / TB, TB, 0, stream>>>(
            t5, st, t5, total, 16, 224 * 224, 1);
    }
    // output conv 3x3 16->1 -> x_tilde
    {
        int total = N * 1 * 224 * 224;
        vq_conv3x3s1<<<(total + TB - 1) / TB, TB, 0, stream>>>(
            t5, w4, b4, x_tilde, N, 16, 1, 224, 224);
    }
}